// MineralDepositGCN_38276748542138
// MI455X (gfx1250) — compile-verified
//
#include <hip/hip_runtime.h>
#include <cstddef>

// MineralDepositGCN on MI455X (gfx1250, wave32).
// GEMMs via V_WMMA_F32_16X16X4_F32 (exact fp32 path; GEMM is <5% of runtime).
// Edge aggregation (the memory-bound part, ~5GB traffic -> ~210us at 23.3TB/s)
// uses one wave per edge with coalesced loads + hardware fp32 global atomics;
// working set (~25MB/buffer) lives in the 192MB L2.

typedef __attribute__((ext_vector_type(2))) float v2f;
typedef __attribute__((ext_vector_type(8))) float v8f;

#define HIDC 64

// ---------------------------------------------------------------------------
// WMMA f32 GEMM: C[M,Nc] = act(A[M,K] @ B[K,Nc] + bias), row-major everywhere.
// One wave computes one 16x16 tile. 4 waves per block cover 64 output columns.
// A layout (16x4 f32): lane(0-15)=rows, VGPR j holds K = k0 + 2*(lane>>4) + j.
// B layout (4x16 f32): lane(0-15)=cols, VGPR j holds K = k0 + 2*(lane>>4) + j.
// C/D layout: VGPR j -> row = row0 + j + 8*(lane>>4), col = n0 + (lane&15).
// K-loop uses pure pointer bumps (no per-iteration 64-bit address math).
// ACT: 0 = none, 1 = exact GELU.
// ---------------------------------------------------------------------------
template <int ACT>
__global__ __launch_bounds__(128) void gemm_wmma_f32(
    const float* __restrict__ A, const float* __restrict__ B,
    const float* __restrict__ bias, float* __restrict__ C,
    int M, int K, int Nc)
{
    const int lane = threadIdx.x & 31;
    const int wave = threadIdx.x >> 5;
    const int hi   = lane >> 4;   // which half of the wave
    const int lo   = lane & 15;
    const int row0 = blockIdx.x << 4;
    const int n0   = (blockIdx.y * 4 + wave) << 4;
    if (n0 >= Nc) return;         // wave-uniform exit (EXEC stays all-ones)

    // Hoist clamped base addresses entirely out of the K loop.
    int arow = row0 + lo; if (arow > M - 1) arow = M - 1;      // clamp tail rows
    int bcol = n0 + lo;   if (bcol > Nc - 1) bcol = Nc - 1;    // clamp tail cols
    const float* ap  = A + (size_t)arow * K + (hi << 1);
    const float* bp0 = B + (size_t)(hi << 1) * Nc + bcol;      // K row k0+2*hi
    const float* bp1 = bp0 + Nc;                               // K row k0+2*hi+1
    const size_t bstep = (size_t)Nc * 4;

    v8f acc = {0.f, 0.f, 0.f, 0.f, 0.f, 0.f, 0.f, 0.f};
    #pragma unroll 4
    for (int k = 0; k < K; k += 4) {
        v2f a = *(const v2f*)ap;                // contiguous float2 per lane
        v2f b;
        b.x = *bp0;
        b.y = *bp1;
        acc = __builtin_amdgcn_wmma_f32_16x16x4_f32(
            /*neg_a=*/false, a, /*neg_b=*/false, b,
            /*c_mod=*/(short)0, acc, /*reuse_a=*/false, /*reuse_b=*/false);
        ap  += 4;
        bp0 += bstep;
        bp1 += bstep;
    }

    const int n = n0 + lo;
    if (n < Nc) {
        const float bv = bias ? bias[n] : 0.f;
        float* cp = C + (size_t)(row0 + (hi << 3)) * Nc + n;
        #pragma unroll
        for (int j = 0; j < 8; ++j) {
            const int m = row0 + j + (hi << 3);
            if (m < M) {
                float v = acc[j] + bv;
                if (ACT == 1)  // exact GELU: 0.5*x*(1+erf(x/sqrt(2)))
                    v = 0.5f * v * (1.f + erff(v * 0.70710678118654752f));
                cp[(size_t)j * Nc] = v;
            }
        }
    }
}

// ---------------------------------------------------------------------------
// Edge scatter: agg[dst[e], :] += h[src[e], :] * ew[e].
// One wave per edge; lane l handles channels l and l+32 (coalesced).
// ---------------------------------------------------------------------------
__global__ __launch_bounds__(256) void gcn_scatter(
    const float* __restrict__ h, const int* __restrict__ src,
    const int* __restrict__ dst, const float* __restrict__ ew,
    float* __restrict__ agg, int E)
{
    const int e = (blockIdx.x * blockDim.x + threadIdx.x) >> 5;
    if (e >= E) return;
    const int lane = threadIdx.x & 31;
    const int s = src[e];
    const int d = dst[e];
    const float w = ew[e];
    const float v0 = h[(size_t)s * HIDC + lane];
    const float v1 = h[(size_t)s * HIDC + lane + 32];
    unsafeAtomicAdd(&agg[(size_t)d * HIDC + lane], v0 * w);
    unsafeAtomicAdd(&agg[(size_t)d * HIDC + lane + 32], v1 * w);
}

// ---------------------------------------------------------------------------
// In-place bias + ReLU, accumulating per-channel sum / sumsq into stats[0:128].
// Each thread owns channel (tid & 63) and strides over rows.
// ---------------------------------------------------------------------------
__global__ __launch_bounds__(256) void bias_relu_stats(
    float* __restrict__ x, const float* __restrict__ bias,
    float* __restrict__ stats, int Nrows)
{
    const int t = blockIdx.x * blockDim.x + threadIdx.x;
    const int c = t & 63;
    const int rstep = (gridDim.x * blockDim.x) >> 6;
    const float b = bias[c];
    float s = 0.f, s2 = 0.f;
    float* p = x + (size_t)(t >> 6) * HIDC + c;
    const size_t pstep = (size_t)rstep * HIDC;
    for (int r = t >> 6; r < Nrows; r += rstep) {
        float v = *p + b;
        v = fmaxf(v, 0.f);
        *p = v;
        s  += v;
        s2 += v * v;
        p  += pstep;
    }
    unsafeAtomicAdd(&stats[c], s);
    unsafeAtomicAdd(&stats[64 + c], s2);
}

// stats[0:64]=sum, [64:128]=sumsq -> [128:192]=scale, [192:256]=shift
__global__ void bn_finalize(float* __restrict__ stats, const float* __restrict__ g,
                            const float* __restrict__ be, float invN)
{
    const int c = threadIdx.x;  // 64 threads
    const float mean = stats[c] * invN;
    const float var  = stats[64 + c] * invN - mean * mean;  // biased variance
    const float sc   = g[c] * rsqrtf(var + 1e-5f);
    stats[128 + c] = sc;
    stats[192 + c] = be[c] - mean * sc;
}

__global__ __launch_bounds__(256) void bn_apply(
    float* __restrict__ x, const float* __restrict__ stats, size_t total)
{
    size_t i = (size_t)blockIdx.x * blockDim.x + threadIdx.x;
    const size_t st = (size_t)gridDim.x * blockDim.x;
    for (; i < total; i += st) {
        const int c = (int)(i & 63);
        x[i] = x[i] * stats[128 + c] + stats[192 + c];
    }
}

__global__ __launch_bounds__(256) void zero_buf4(float4* __restrict__ p, size_t n4)
{
    size_t i = (size_t)blockIdx.x * blockDim.x + threadIdx.x;
    const size_t st = (size_t)gridDim.x * blockDim.x;
    const float4 z = make_float4(0.f, 0.f, 0.f, 0.f);
    for (; i < n4; i += st) p[i] = z;
}

extern "C" void kernel_launch(void* const* d_in, const int* in_sizes, int n_in,
                              void* d_out, int out_size, void* d_ws, size_t ws_size,
                              hipStream_t stream)
{
    const float* x  = (const float*)d_in[0];
    const int*   ei = (const int*)d_in[1];
    const float* ew = (const float*)d_in[2];
    const float* cw[3] = {(const float*)d_in[3], (const float*)d_in[5], (const float*)d_in[7]};
    const float* cb[3] = {(const float*)d_in[4], (const float*)d_in[6], (const float*)d_in[8]};
    const float* bg[3] = {(const float*)d_in[9],  (const float*)d_in[11], (const float*)d_in[13]};
    const float* bb[3] = {(const float*)d_in[10], (const float*)d_in[12], (const float*)d_in[14]};
    const float* w1 = (const float*)d_in[15]; const float* b1 = (const float*)d_in[16];
    const float* w2 = (const float*)d_in[17]; const float* b2 = (const float*)d_in[18];
    const float* w3 = (const float*)d_in[19]; const float* b3 = (const float*)d_in[20];

    const int N = in_sizes[0] / 128;   // nodes
    const int E = in_sizes[2];         // edges
    const int* src = ei;               // edge_index[0]
    const int* dst = ei + E;           // edge_index[1]

    // Workspace layout (floats): bufA N*128 | bufB N*64 | stats 256  (~77 MB)
    float* bufA  = (float*)d_ws;
    float* bufB  = bufA + (size_t)N * 128;
    float* stats = bufB + (size_t)N * 64;

    const dim3 blk128(128), blk256(256);
    const int rowTiles = (N + 15) / 16;
    const int eBlocks  = (E * 32 + 255) / 256;  // one wave per edge

    const float* in_ptr = x;
    int K = 128;
    for (int l = 0; l < 3; ++l) {
        // h = in @ W   (no bias; GCN bias applied after aggregation)
        gemm_wmma_f32<0><<<dim3(rowTiles, 1), blk128, 0, stream>>>(
            in_ptr, cw[l], nullptr, bufA, N, K, HIDC);
        // zero agg + stats (graph-replay safe)
        zero_buf4<<<2048, blk256, 0, stream>>>((float4*)bufB, (size_t)N * HIDC / 4);
        zero_buf4<<<1, 64, 0, stream>>>((float4*)stats, 64);
        // agg[dst] += h[src] * ew
        gcn_scatter<<<eBlocks, blk256, 0, stream>>>(bufA, src, dst, ew, bufB, E);
        // relu(agg + b), accumulate batch stats (in place)
        bias_relu_stats<<<1024, blk256, 0, stream>>>(bufB, cb[l], stats, N);
        // fold mean/var into scale/shift, then apply BN in place
        bn_finalize<<<1, 64, 0, stream>>>(stats, bg[l], bb[l], 1.0f / (float)N);
        bn_apply<<<2048, blk256, 0, stream>>>(bufB, stats, (size_t)N * HIDC);
        in_ptr = bufB;
        K = HIDC;
    }

    // classify MLP: 64 -> 128 (gelu) -> 64 (gelu) -> 5
    gemm_wmma_f32<1><<<dim3(rowTiles, 2), blk128, 0, stream>>>(bufB, w1, b1, bufA, N, 64, 128);
    gemm_wmma_f32<1><<<dim3(rowTiles, 1), blk128, 0, stream>>>(bufA, w2, b2, bufB, N, 128, 64);
    gemm_wmma_f32<0><<<dim3(rowTiles, 1), blk128, 0, stream>>>(bufB, w3, b3, (float*)d_out, N, 64, 5);
    (void)n_in; (void)out_size; (void)ws_size;
}